// IrrepsLinear_89979564851605
// MI455X (gfx1250) — compile-verified
//
#include <hip/hip_runtime.h>

typedef __attribute__((ext_vector_type(2))) float v2f;
typedef __attribute__((ext_vector_type(8))) float v8f;

#define ROWS 32          // rows of x per workgroup (two 16-row WMMA n-tiles)
#define NCOL 592         // features in/out
#define LDSW 593         // padded LDS row stride (17 mod 64 banks -> conflict free)
#define NWAVES 8
#define NCOMBO 37        // (l, d, otile) tasks: 8 + 12 + 10 + 7

__global__ __launch_bounds__(256)
void irreps_linear_wmma(const float* __restrict__ x,
                        const float* __restrict__ w0, const float* __restrict__ w1,
                        const float* __restrict__ w2, const float* __restrict__ w3,
                        const float* __restrict__ b0, const float* __restrict__ b1,
                        const float* __restrict__ b2, const float* __restrict__ b3,
                        float* __restrict__ out, int N)
{
    __shared__ float lds_x[ROWS * LDSW];
    __shared__ float lds_o[ROWS * LDSW];

    const int tid      = threadIdx.x;
    const int lane     = tid & 31;            // wave32
    const int row_base = blockIdx.x * ROWS;

    // wave id is wave-uniform: force it scalar so the whole combo decode,
    // loop bounds and predicates live in SGPRs (s_cbranch, not exec masking)
    const int wave = __builtin_amdgcn_readfirstlane(tid >> 5);   // 0..7

    // ---- stage 32 rows of x into LDS (coalesced global reads) ----
    for (int idx = tid; idx < ROWS * NCOL; idx += 256) {
        int r = idx / NCOL, c = idx - r * NCOL;
        int gr = row_base + r;
        lds_x[r * LDSW + c] = (gr < N) ? x[(size_t)gr * NCOL + c] : 0.0f;
    }
    __syncthreads();

    const int laneM = lane & 15;          // column (output channel) / A-row index
    const int kkb   = (lane >> 4) * 2;    // K sub-offset per ISA 32-bit A/B layout
    const int mOff  = (lane >> 4) * 8;    // D-matrix M offset per 32-bit C/D layout

    for (int combo = wave; combo < NCOMBO; combo += NWAVES) {
        // decode (l, d, otile) and per-l parameters -- all scalar
        int K, dl, seg, ot, d;
        const float* w; const float* bv;
        if (combo < 8) {
            K = 128; dl = 1; seg = 0;   w = w0; bv = b0; ot = combo;            d = 0;
        } else if (combo < 20) {
            K = 64;  dl = 3; seg = 128; w = w1; bv = b1; ot = (combo - 8) & 3;  d = (combo - 8) >> 2;
        } else if (combo < 30) {
            K = 32;  dl = 5; seg = 320; w = w2; bv = b2; ot = (combo - 20) & 1; d = (combo - 20) >> 1;
        } else {
            K = 16;  dl = 7; seg = 480; w = w3; bv = b3; ot = 0;                d = combo - 30;
        }

        const int   o    = ot * 16 + laneM;       // output channel this lane owns in B/D
        const float bias = bv[o];

        const float* lxa0 = &lds_x[laneM * LDSW + seg + d];          // n-tile 0 rows
        const float* lxa1 = &lds_x[(16 + laneM) * LDSW + seg + d];   // n-tile 1 rows
        const float* wrow = &w[o * K];                               // B[k][o] = w[o][k]

        v8f acc0 = {}; v8f acc1 = {};
        #pragma unroll 4
        for (int k0 = 0; k0 < K; k0 += 4) {
            const int k = k0 + kkb;
            v2f a0, a1, b;
            // A (16x4 f32): lanes0-15 hold K={0,1}, lanes16-31 hold K={2,3}
            a0.x = lxa0[(k    ) * dl];
            a0.y = lxa0[(k + 1) * dl];
            a1.x = lxa1[(k    ) * dl];
            a1.y = lxa1[(k + 1) * dl];
            // B (4x16 f32): mirrored layout, column = laneM
            b.x  = wrow[k];
            b.y  = wrow[k + 1];
            acc0 = __builtin_amdgcn_wmma_f32_16x16x4_f32(false, a0, false, b,
                                                         (short)0, acc0, false, false);
            acc1 = __builtin_amdgcn_wmma_f32_16x16x4_f32(false, a1, false, b,
                                                         (short)0, acc1, false, false);
        }

        // D (16x16 f32): VGPR g -> row M = mOff + g, col N = laneM
        const int ocol = seg + o * dl + d;
        #pragma unroll
        for (int g = 0; g < 8; ++g) {
            float v0 = acc0[g] + bias;
            float v1 = acc1[g] + bias;
            if (K == 128) {  // l == 0 (scalar branch): smooth leaky relu, alpha = 0.2
                // sigma via v_exp_f32 + v_rcp_f32 (TRANS32) instead of IEEE divide
                float s0 = __builtin_amdgcn_rcpf(1.0f + __expf(-v0));
                float s1 = __builtin_amdgcn_rcpf(1.0f + __expf(-v1));
                v0 = 0.6f * v0 + 0.4f * v0 * (2.0f * s0 - 1.0f);
                v1 = 0.6f * v1 + 0.4f * v1 * (2.0f * s1 - 1.0f);
            }
            const int nl = mOff + g;
            lds_o[nl * LDSW + ocol]        = v0;
            lds_o[(16 + nl) * LDSW + ocol] = v1;
        }
    }
    __syncthreads();

    // ---- coalesced write-back of the 32x592 output tile ----
    for (int idx = tid; idx < ROWS * NCOL; idx += 256) {
        int r = idx / NCOL, c = idx - r * NCOL;
        int gr = row_base + r;
        if (gr < N) out[(size_t)gr * NCOL + c] = lds_o[r * LDSW + c];
    }
}

extern "C" void kernel_launch(void* const* d_in, const int* in_sizes, int n_in,
                              void* d_out, int out_size, void* d_ws, size_t ws_size,
                              hipStream_t stream) {
    const float* x  = (const float*)d_in[0];
    const float* w0 = (const float*)d_in[1];
    const float* w1 = (const float*)d_in[2];
    const float* w2 = (const float*)d_in[3];
    const float* w3 = (const float*)d_in[4];
    const float* b0 = (const float*)d_in[5];
    const float* b1 = (const float*)d_in[6];
    const float* b2 = (const float*)d_in[7];
    const float* b3 = (const float*)d_in[8];
    float* out = (float*)d_out;

    const int N = in_sizes[0] / NCOL;   // 100000
    const int grid = (N + ROWS - 1) / ROWS;
    irreps_linear_wmma<<<grid, 256, 0, stream>>>(x, w0, w1, w2, w3,
                                                 b0, b1, b2, b3, out, N);
}